// GIN_55800215109866
// MI455X (gfx1250) — compile-verified
//
#include <hip/hip_runtime.h>
#include <hip/hip_bf16.h>

#define N_NODES  100000
#define N_EDGES  1600000
#define N_GRAPHS 512
#define F_IN     128
#define HDIM     64
#define L_LAYERS 3
#define HS_COLS  (L_LAYERS * HDIM)      // 192
#define G_COLS   (3 * HS_COLS)          // 576

typedef __attribute__((ext_vector_type(2))) float v2f;
typedef __attribute__((ext_vector_type(8))) float v8f;

// ---------------------------------------------------------------------------
// WMMA helper: D = A(16x4,f32) * B(4x16,f32) + C  (exact fp32 matrix op)
// ---------------------------------------------------------------------------
__device__ __forceinline__ v8f wmma_f32_k4(v2f a, v2f b, v8f c) {
    return __builtin_amdgcn_wmma_f32_16x16x4_f32(
        /*neg_a=*/false, a, /*neg_b=*/false, b,
        /*c_mod=*/(short)0, c, /*reuse_a=*/false, /*reuse_b=*/false);
}

// ---------------------------------------------------------------------------
// Edge-parallel scatter-add:  agg[dst] += x[src]   (float4 per thread)
// grid: (ceil(E/256), K/4), block 256
// ---------------------------------------------------------------------------
__global__ void gin_scatter_kernel(const float* __restrict__ x, int xstride,
                                   const long long* __restrict__ src,
                                   const long long* __restrict__ dst,
                                   float* __restrict__ agg, int K) {
    long long e = (long long)blockIdx.x * blockDim.x + threadIdx.x;
    if (e >= N_EDGES) return;
    int f4 = blockIdx.y * 4;
    long long s = src[e];
    long long d = dst[e];
    const float4 v = *(const float4*)(x + s * (long long)xstride + f4);
    float* o = agg + d * (long long)K + f4;
    unsafeAtomicAdd(o + 0, v.x);
    unsafeAtomicAdd(o + 1, v.y);
    unsafeAtomicAdd(o + 2, v.z);
    unsafeAtomicAdd(o + 3, v.w);
}

// ---------------------------------------------------------------------------
// One GIN layer MLP via fp32 WMMA.
// One wave32 per block computes 16 nodes x 64 features:
//   h  = (1+eps)*x + agg
//   h1 = relu(h @ w1 + b1)         (K x 64 GEMM, K = 128 or 64)
//   h2 = relu(h1 @ w2 + b2)        (64 x 64 GEMM, staged through LDS)
// ---------------------------------------------------------------------------
__global__ __launch_bounds__(32)
void gin_mlp_kernel(const float* __restrict__ x, int xstride,
                    const float* __restrict__ agg,
                    const float* __restrict__ eps,    // points at eps[layer]
                    const float* __restrict__ w1,     // [K,64]
                    const float* __restrict__ b1,
                    const float* __restrict__ w2,     // [64,64]
                    const float* __restrict__ b2,
                    float* __restrict__ out, int ostride,
                    int K) {
    __shared__ float lds_h[16 * HDIM];

    const int lane  = threadIdx.x;
    const int row   = lane & 15;          // A row / B column (ISA 7.12.2)
    const int khalf = (lane >> 4) * 2;    // lanes 16-31 hold K+2,K+3
    const int node0 = blockIdx.x * 16;
    const float e1  = 1.0f + eps[0];

    const float* xr = x   + (size_t)(node0 + row) * xstride;
    const float* ar = agg + (size_t)(node0 + row) * K;

    v8f c0 = {}, c1 = {}, c2 = {}, c3 = {};
    for (int k0 = 0; k0 < K; k0 += 4) {
        const int ka = k0 + khalf;
        v2f a;
        a.x = e1 * xr[ka]     + ar[ka];
        a.y = e1 * xr[ka + 1] + ar[ka + 1];
        const float* wr0 = w1 + (size_t)ka * HDIM + row;
        const float* wr1 = wr0 + HDIM;
        v2f b;
        b.x = wr0[0];  b.y = wr1[0];   c0 = wmma_f32_k4(a, b, c0);
        b.x = wr0[16]; b.y = wr1[16];  c1 = wmma_f32_k4(a, b, c1);
        b.x = wr0[32]; b.y = wr1[32];  c2 = wmma_f32_k4(a, b, c2);
        b.x = wr0[48]; b.y = wr1[48];  c3 = wmma_f32_k4(a, b, c3);
    }

    // bias + relu -> LDS (D layout: VGPR r = rows r and r+8)
    const int drow = (lane >> 4) * 8;
    const int dcol = lane & 15;
#pragma unroll
    for (int r = 0; r < 8; ++r) {
        lds_h[(drow + r) * HDIM + dcol +  0] = fmaxf(c0[r] + b1[dcol +  0], 0.0f);
        lds_h[(drow + r) * HDIM + dcol + 16] = fmaxf(c1[r] + b1[dcol + 16], 0.0f);
        lds_h[(drow + r) * HDIM + dcol + 32] = fmaxf(c2[r] + b1[dcol + 32], 0.0f);
        lds_h[(drow + r) * HDIM + dcol + 48] = fmaxf(c3[r] + b1[dcol + 48], 0.0f);
    }
    __syncthreads();

    // second GEMM: h2 = relu(h1 @ w2 + b2), K = 64, A from LDS
    v8f d0 = {}, d1 = {}, d2 = {}, d3 = {};
    for (int k0 = 0; k0 < HDIM; k0 += 4) {
        const int ka = k0 + khalf;
        v2f a;
        a.x = lds_h[row * HDIM + ka];
        a.y = lds_h[row * HDIM + ka + 1];
        const float* wr0 = w2 + (size_t)ka * HDIM + row;
        const float* wr1 = wr0 + HDIM;
        v2f b;
        b.x = wr0[0];  b.y = wr1[0];   d0 = wmma_f32_k4(a, b, d0);
        b.x = wr0[16]; b.y = wr1[16];  d1 = wmma_f32_k4(a, b, d1);
        b.x = wr0[32]; b.y = wr1[32];  d2 = wmma_f32_k4(a, b, d2);
        b.x = wr0[48]; b.y = wr1[48];  d3 = wmma_f32_k4(a, b, d3);
    }
#pragma unroll
    for (int r = 0; r < 8; ++r) {
        float* orow = out + (size_t)(node0 + drow + r) * ostride + dcol;
        orow[ 0] = fmaxf(d0[r] + b2[dcol +  0], 0.0f);
        orow[16] = fmaxf(d1[r] + b2[dcol + 16], 0.0f);
        orow[32] = fmaxf(d2[r] + b2[dcol + 32], 0.0f);
        orow[48] = fmaxf(d3[r] + b2[dcol + 48], 0.0f);
    }
}

// ---------------------------------------------------------------------------
// Per-graph node count (as float, for mean)
// ---------------------------------------------------------------------------
__global__ void gin_count_kernel(const long long* __restrict__ batch,
                                 float* __restrict__ cnt) {
    int n = blockIdx.x * blockDim.x + threadIdx.x;
    if (n >= N_NODES) return;
    unsafeAtomicAdd(&cnt[batch[n]], 1.0f);
}

// ---------------------------------------------------------------------------
// Pool: sum + max over batch segments.  hs values are post-ReLU (>= 0) so
// float max == uint max on the bit pattern.  grid: (ceil(N/256), 192/4)
// ---------------------------------------------------------------------------
__global__ void gin_pool_kernel(const float* __restrict__ hs,
                                const long long* __restrict__ batch,
                                float* __restrict__ psum,
                                unsigned int* __restrict__ pmax) {
    int n = blockIdx.x * blockDim.x + threadIdx.x;
    if (n >= N_NODES) return;
    int f4 = blockIdx.y * 4;
    long long g = batch[n];
    const float4 v = *(const float4*)(hs + (long long)n * HS_COLS + f4);
    float* s = psum + g * HS_COLS + f4;
    unsafeAtomicAdd(s + 0, v.x);
    unsafeAtomicAdd(s + 1, v.y);
    unsafeAtomicAdd(s + 2, v.z);
    unsafeAtomicAdd(s + 3, v.w);
    unsigned int* m = pmax + g * HS_COLS + f4;
    atomicMax(m + 0, __float_as_uint(v.x));
    atomicMax(m + 1, __float_as_uint(v.y));
    atomicMax(m + 2, __float_as_uint(v.z));
    atomicMax(m + 3, __float_as_uint(v.w));
}

// ---------------------------------------------------------------------------
// Head: g = [mean | max | sum] (512 x 576) -> fc1(576,64)+relu -> fc2(64,2)
// -> sigmoid.  One wave32 per 16 graphs; fc1 via WMMA, fc2 per-lane.
// ---------------------------------------------------------------------------
__device__ __forceinline__ float g_feat(const float* psum,
                                        const unsigned int* pmax,
                                        int g, float rcnt, int j) {
    if (j < HS_COLS)       return psum[g * HS_COLS + j] * rcnt;            // mean
    if (j < 2 * HS_COLS)   return __uint_as_float(pmax[g * HS_COLS + j - HS_COLS]); // max
    return psum[g * HS_COLS + j - 2 * HS_COLS];                            // sum
}

__global__ __launch_bounds__(32)
void gin_head_kernel(const float* __restrict__ psum,
                     const unsigned int* __restrict__ pmax,
                     const float* __restrict__ cnt,
                     const float* __restrict__ fc1w,   // [576,64]
                     const float* __restrict__ fc1b,
                     const float* __restrict__ fc2w,   // [64,2]
                     const float* __restrict__ fc2b,
                     float* __restrict__ out) {        // [512,2]
    __shared__ float lds_h[16 * HDIM];

    const int lane  = threadIdx.x;
    const int row   = lane & 15;
    const int khalf = (lane >> 4) * 2;
    const int g0    = blockIdx.x * 16;
    const int g     = g0 + row;
    const float rc  = 1.0f / fmaxf(cnt[g], 1.0f);

    v8f c0 = {}, c1 = {}, c2 = {}, c3 = {};
    for (int k0 = 0; k0 < G_COLS; k0 += 4) {
        const int ka = k0 + khalf;
        v2f a;
        a.x = g_feat(psum, pmax, g, rc, ka);
        a.y = g_feat(psum, pmax, g, rc, ka + 1);
        const float* wr0 = fc1w + (size_t)ka * HDIM + row;
        const float* wr1 = wr0 + HDIM;
        v2f b;
        b.x = wr0[0];  b.y = wr1[0];   c0 = wmma_f32_k4(a, b, c0);
        b.x = wr0[16]; b.y = wr1[16];  c1 = wmma_f32_k4(a, b, c1);
        b.x = wr0[32]; b.y = wr1[32];  c2 = wmma_f32_k4(a, b, c2);
        b.x = wr0[48]; b.y = wr1[48];  c3 = wmma_f32_k4(a, b, c3);
    }

    const int drow = (lane >> 4) * 8;
    const int dcol = lane & 15;
#pragma unroll
    for (int r = 0; r < 8; ++r) {
        lds_h[(drow + r) * HDIM + dcol +  0] = fmaxf(c0[r] + fc1b[dcol +  0], 0.0f);
        lds_h[(drow + r) * HDIM + dcol + 16] = fmaxf(c1[r] + fc1b[dcol + 16], 0.0f);
        lds_h[(drow + r) * HDIM + dcol + 32] = fmaxf(c2[r] + fc1b[dcol + 32], 0.0f);
        lds_h[(drow + r) * HDIM + dcol + 48] = fmaxf(c3[r] + fc1b[dcol + 48], 0.0f);
    }
    __syncthreads();

    // fc2 + sigmoid: lane -> (row = lane>>1, class = lane&1); 16x2 = 32 outputs
    const int orow = lane >> 1;
    const int ocol = lane & 1;
    float acc = fc2b[ocol];
#pragma unroll 8
    for (int k = 0; k < HDIM; ++k)
        acc += lds_h[orow * HDIM + k] * fc2w[k * 2 + ocol];
    out[(g0 + orow) * 2 + ocol] = 1.0f / (1.0f + __expf(-acc));
}

// ---------------------------------------------------------------------------
// Host-side orchestration
// ---------------------------------------------------------------------------
extern "C" void kernel_launch(void* const* d_in, const int* in_sizes, int n_in,
                              void* d_out, int out_size, void* d_ws, size_t ws_size,
                              hipStream_t stream) {
    const float*     x     = (const float*)d_in[0];
    const long long* ei    = (const long long*)d_in[1];   // [2, E] int64
    const long long* batch = (const long long*)d_in[2];   // [N] int64
    const float* lw1[3] = { (const float*)d_in[3], (const float*)d_in[7],  (const float*)d_in[11] };
    const float* lb1[3] = { (const float*)d_in[4], (const float*)d_in[8],  (const float*)d_in[12] };
    const float* lw2[3] = { (const float*)d_in[5], (const float*)d_in[9],  (const float*)d_in[13] };
    const float* lb2[3] = { (const float*)d_in[6], (const float*)d_in[10], (const float*)d_in[14] };
    const float* eps  = (const float*)d_in[15];
    const float* fc1w = (const float*)d_in[16];
    const float* fc1b = (const float*)d_in[17];
    const float* fc2w = (const float*)d_in[18];
    const float* fc2b = (const float*)d_in[19];

    const long long* src = ei;
    const long long* dst = ei + N_EDGES;

    // workspace carve-up (256B aligned)
    char* ws = (char*)d_ws;
    size_t off = 0;
    auto carve = [&](size_t bytes) {
        char* p = ws + off;
        off = (off + bytes + 255) & ~(size_t)255;
        return p;
    };
    float*        agg  = (float*)carve((size_t)N_NODES * F_IN * sizeof(float));
    float*        hs   = (float*)carve((size_t)N_NODES * HS_COLS * sizeof(float));
    float*        psum = (float*)carve((size_t)N_GRAPHS * HS_COLS * sizeof(float));
    unsigned int* pmax = (unsigned int*)carve((size_t)N_GRAPHS * HS_COLS * sizeof(unsigned int));
    float*        cnt  = (float*)carve((size_t)N_GRAPHS * sizeof(float));
    (void)ws_size;

    const int eblocks = (N_EDGES + 255) / 256;   // 6250
    const int nblocks = (N_NODES + 255) / 256;   // 391
    const int tblocks = N_NODES / 16;            // 6250 (exact)

    for (int i = 0; i < L_LAYERS; ++i) {
        const int K = (i == 0) ? F_IN : HDIM;
        const float* xin = (i == 0) ? x : (hs + (size_t)(i - 1) * HDIM);
        const int xstride = (i == 0) ? F_IN : HS_COLS;

        hipMemsetAsync(agg, 0, (size_t)N_NODES * K * sizeof(float), stream);
        gin_scatter_kernel<<<dim3(eblocks, K / 4), 256, 0, stream>>>(
            xin, xstride, src, dst, agg, K);
        gin_mlp_kernel<<<tblocks, 32, 0, stream>>>(
            xin, xstride, agg, eps + i, lw1[i], lb1[i], lw2[i], lb2[i],
            hs + (size_t)i * HDIM, HS_COLS, K);
    }

    hipMemsetAsync(psum, 0, (size_t)N_GRAPHS * HS_COLS * sizeof(float), stream);
    hipMemsetAsync(pmax, 0, (size_t)N_GRAPHS * HS_COLS * sizeof(unsigned int), stream);
    hipMemsetAsync(cnt,  0, (size_t)N_GRAPHS * sizeof(float), stream);

    gin_count_kernel<<<nblocks, 256, 0, stream>>>(batch, cnt);
    gin_pool_kernel<<<dim3(nblocks, HS_COLS / 4), 256, 0, stream>>>(hs, batch, psum, pmax);
    gin_head_kernel<<<N_GRAPHS / 16, 32, 0, stream>>>(
        psum, pmax, cnt, fc1w, fc1b, fc2w, fc2b, (float*)d_out);
}